// NetVLAD_1580547971686
// MI455X (gfx1250) — compile-verified
//
#include <hip/hip_runtime.h>
#include <hip/hip_bf16.h>
#include <cstdint>

// ---------------- problem constants (match reference) ----------------
#define NN   32
#define CC   512
#define TT   300
#define RR   4
#define TRR  1200      // T*R
#define KK   64
#define OUTD 1024
#define KC   (KK*CC)   // 32768
#define EPSN 1e-12f
#define EPSL 1e-5f

typedef unsigned int u32t;
typedef __attribute__((ext_vector_type(2))) float v2f;
typedef __attribute__((ext_vector_type(8))) float v8f;
typedef __attribute__((ext_vector_type(4))) u32t  v4u;
typedef __attribute__((ext_vector_type(8))) int   v8i;
typedef __attribute__((ext_vector_type(4))) int   v4i;

union V8 { v8f v; float f[8]; };

// D = A(16x4,f32) * B(4x16,f32) + C(16x16,f32)  -- CDNA5 V_WMMA_F32_16X16X4_F32
__device__ __forceinline__ v8f wmma4(v2f a, v2f b, v8f c) {
  return __builtin_amdgcn_wmma_f32_16x16x4_f32(
      /*neg_a=*/false, a, /*neg_b=*/false, b,
      /*c_mod=*/(short)0, c, /*reuse_a=*/false, /*reuse_b=*/false);
}

// ---------------------------------------------------------------------
// Tensor Data Mover: 2-D strided tile (f32 elements) global -> LDS.
// D# per CDNA5 ISA; optional 1-DWORD row pad
// every 2^(pad_interval+1) DWORDs keeps LDS rows bank-conflict free.
// Issue from one wave; wait with s_wait_tensorcnt.
// amdgpu-toolchain (clang-23) 6-arg builtin form.
// ---------------------------------------------------------------------
__device__ __forceinline__ void tdm_load_2d(
    const void* g, void* lds,
    u32t tile_cols, u32t tile_rows,
    u32t tensor_cols, u32t tensor_rows,
    u32t row_stride_elems, u32t pad_interval /*0 = no pad*/)
{
  const unsigned long long ga = (unsigned long long)(uintptr_t)g;
  v4u g0;
  g0.x = 1u;                                            // count=1, user D#
  g0.y = (u32t)(uintptr_t)lds;                          // lds_addr
  g0.z = (u32t)ga;                                      // global_addr[31:0]
  g0.w = (u32t)((ga >> 32) & 0x01FFFFFFu) | (2u << 30); // addr[56:32] | type=2
  u32t w0 = (2u << 16);                                 // data_size = 4B
  if (pad_interval) w0 |= (1u << 20) | (pad_interval << 22); // pad_en, pad_amount=0 (1 dw)
  v8i g1;
  g1[0] = (int)w0;
  g1[1] = (int)((tensor_cols & 0xFFFFu) << 16);                               // tensor_dim0 lo
  g1[2] = (int)(((tensor_cols >> 16) & 0xFFFFu) | ((tensor_rows & 0xFFFFu) << 16));
  g1[3] = (int)(((tensor_rows >> 16) & 0xFFFFu) | ((tile_cols & 0xFFFFu) << 16));
  g1[4] = (int)(tile_rows & 0xFFFFu);                                         // tile_dim1 (tile_dim2=0)
  g1[5] = (int)row_stride_elems;                                              // tensor_dim0_stride[31:0]
  g1[6] = 0;
  g1[7] = 0;
  v4i z4 = {};
  v8i z8 = {};
  __builtin_amdgcn_tensor_load_to_lds(g0, g1, z4, z4, z8, 0);
}

// Async per-lane 16B copy global -> LDS (ASYNCcnt path)
__device__ __forceinline__ void a_g2l_b128(void* lds, const void* g) {
  const u32t l = (u32t)(uintptr_t)lds;
  asm volatile("global_load_async_to_lds_b128 %0, %1, off"
               :: "v"(l), "v"(g) : "memory");
}
__device__ __forceinline__ void a_wait0() {
  asm volatile("s_wait_asynccnt 0x0" ::: "memory");
}

// =====================================================================
// Kernel 1: logits = conv_w @ x  (per n, per 16-col t tile), softmax(K),
//           mask, write am (N,K,TR).  x slab staged once by TDM.
// =====================================================================
__global__ __launch_bounds__(128) void k_assign(
    const float* __restrict__ x, const int* __restrict__ mask,
    const float* __restrict__ conv_w, float* __restrict__ am)
{
  const int n   = blockIdx.y;
  const int tb  = blockIdx.x * 16;            // t-tile base (75 tiles)
  const int tid = threadIdx.x;
  const int w    = tid >> 5;
  const int lane = tid & 31;
  const int r16  = lane & 15;
  const int hi   = lane >> 4;
  const int kb   = w * 16;

  __shared__ float xs[512][17];               // x slab (TDM pads every 16 dw)
  __shared__ float ls[64][17];                // logits / am staging

  if (tid < 32) {
    tdm_load_2d(x + (size_t)n * CC * TRR + tb, &xs[0][0],
                /*tile*/16, 512, /*tensor*/TRR, CC, TRR, /*pad@16dw*/3);
    __builtin_amdgcn_s_wait_tensorcnt((short)0);
  }
  __syncthreads();

  v8f acc = {};
  const float* cw = conv_w + (size_t)(kb + r16) * CC;
  for (int c = 0; c < CC; c += 4) {
    v2f a, b;
    a.x = cw[c + 2*hi];                        // A[M=k][K=c]
    a.y = cw[c + 2*hi + 1];
    b.x = xs[c + 2*hi][r16];                   // B[K=c][N=t]
    b.y = xs[c + 2*hi + 1][r16];
    acc = wmma4(a, b, acc);
  }
  {
    V8 rr; rr.v = acc;
    #pragma unroll
    for (int i = 0; i < 8; ++i) ls[kb + i + 8*hi][r16] = rr.f[i];
  }
  __syncthreads();

  if (tid < 16) {                              // softmax over K per column
    const int j = tid;
    const float mval = (float)mask[n * TT + (tb + j) / RR];
    float mx = -3.4e38f;
    for (int k = 0; k < KK; ++k) mx = fmaxf(mx, ls[k][j]);
    float s = 0.f;
    for (int k = 0; k < KK; ++k) { float e = __expf(ls[k][j] - mx); ls[k][j] = e; s += e; }
    const float sc = mval / s;
    for (int k = 0; k < KK; ++k) ls[k][j] *= sc;
  }
  __syncthreads();

  for (int idx = tid; idx < KK * 16; idx += 128) {
    const int k = idx >> 4, j = idx & 15;
    am[((size_t)n * KK + k) * TRR + tb + j] = ls[k][j];
  }
}

// =====================================================================
// Kernel 1b: asum[n,k] = sum_t am[n,k,t]  (deterministic tree reduce)
// =====================================================================
__global__ __launch_bounds__(128) void k_asum(
    const float* __restrict__ am, float* __restrict__ asum)
{
  const int k = blockIdx.x, n = blockIdx.y, tid = threadIdx.x;
  __shared__ float red[128];
  const float* row = am + ((size_t)n * KK + k) * TRR;
  float s = 0.f;
  for (int t = tid; t < TRR; t += 128) s += row[t];
  red[tid] = s; __syncthreads();
  for (int off = 64; off > 0; off >>= 1) {
    if (tid < off) red[tid] += red[tid + off];
    __syncthreads();
  }
  if (tid == 0) asum[n * KK + k] = red[0];
}

// =====================================================================
// Kernel 2: vlad[n,k,c] = sum_t am[n,k,t]*x[n,c,t] - asum*centroid
//   x slab via TDM, am tile via async b128 (ASYNCcnt).
// =====================================================================
__global__ __launch_bounds__(256) void k_vlad(
    const float* __restrict__ x, const float* __restrict__ am,
    const float* __restrict__ asum, const float* __restrict__ centroids,
    float* __restrict__ vlad)
{
  const int n   = blockIdx.y;
  const int cb  = blockIdx.x * 128;
  const int tid = threadIdx.x;
  const int w = tid >> 5, lane = tid & 31, r16 = lane & 15, hi = lane >> 4;
  const int kb = (w & 3) * 16;
  const int ch = (w >> 2) * 64;

  __shared__ float xs[128][17];   // TDM pad every 16 dw
  __shared__ float as[64][19];    // pad 19: 76 B rows (not 16B-aligned), gcd(19,64)=1

  const int arow = tid >> 2, ac0 = (tid & 3) * 4;   // am staging map (b128)

  v8f acc[4] = {};

  for (int tc = 0; tc < TRR; tc += 16) {
    a_g2l_b128(&as[arow][ac0],
               am + ((size_t)n * KK + arow) * TRR + tc + ac0);
    if (tid < 32)
      tdm_load_2d(x + ((size_t)n * CC + cb) * TRR + tc, &xs[0][0],
                  16, 128, TRR, CC, TRR, 3);
    a_wait0();
    if (tid < 32) __builtin_amdgcn_s_wait_tensorcnt((short)0);
    __syncthreads();

    #pragma unroll
    for (int ts = 0; ts < 16; ts += 4) {
      v2f a;
      a.x = as[kb + r16][ts + 2*hi];            // A[M=k][Kdim=t]
      a.y = as[kb + r16][ts + 2*hi + 1];
      #pragma unroll
      for (int jt = 0; jt < 4; ++jt) {
        v2f b;                                   // B[Kdim=t][N=c] = x[c][t]
        b.x = xs[ch + jt*16 + r16][ts + 2*hi];
        b.y = xs[ch + jt*16 + r16][ts + 2*hi + 1];
        acc[jt] = wmma4(a, b, acc[jt]);
      }
    }
    __syncthreads();
  }

  #pragma unroll
  for (int jt = 0; jt < 4; ++jt) {
    V8 rr; rr.v = acc[jt];
    #pragma unroll
    for (int i = 0; i < 8; ++i) {
      const int k = kb + i + 8*hi;
      const int c = cb + ch + jt*16 + r16;
      const float val = rr.f[i] - asum[n * KK + k] * centroids[(size_t)k * CC + c];
      vlad[((size_t)n * KK + k) * CC + c] = val;
    }
  }
}

// =====================================================================
// Kernel 3: per-(n,k) intra L2 norm + per-n global L2 norm, folded into
//           one in-place scale pass:  v *= inv[k] * gsc[n]
// =====================================================================
__global__ __launch_bounds__(256) void k_norm(float* __restrict__ vlad)
{
  const int n = blockIdx.x, tid = threadIdx.x;
  __shared__ float red[256];
  __shared__ float invk[KK];
  __shared__ float gsh;
  float g = 0.f;
  for (int k = 0; k < KK; ++k) {
    const float* row = vlad + ((size_t)n * KK + k) * CC;
    const float v0 = row[tid], v1 = row[tid + 256];
    red[tid] = v0 * v0 + v1 * v1;
    __syncthreads();
    for (int off = 128; off > 0; off >>= 1) {
      if (tid < off) red[tid] += red[tid + off];
      __syncthreads();
    }
    if (tid == 0) {
      const float ss  = red[0];
      const float inv = 1.f / fmaxf(sqrtf(ss), EPSN);
      invk[k] = inv;
      g += ss * inv * inv;                       // ||normalized row||^2
    }
    __syncthreads();
  }
  if (tid == 0) gsh = 1.f / fmaxf(sqrtf(g), EPSN);
  __syncthreads();
  const float gsc = gsh;
  float* base = vlad + (size_t)n * KC;
  for (int idx = tid; idx < KC; idx += 256) {
    const int k = idx >> 9;                      // idx / 512
    base[idx] *= invk[k] * gsc;
  }
}

// =====================================================================
// Kernel 4: y = vnorm @ red_w^T   (32 x 32768 x 1024), tiles via TDM
// =====================================================================
__global__ __launch_bounds__(256) void k_reduce(
    const float* __restrict__ vnorm, const float* __restrict__ red_w,
    float* __restrict__ y)
{
  const int ob  = blockIdx.x * 64;
  const int tid = threadIdx.x;
  const int w = tid >> 5, lane = tid & 31, r16 = lane & 15, hi = lane >> 4;
  const int mt = (w & 1) * 16, ot = (w >> 1) * 16;

  __shared__ float rws[64][33];   // TDM pad every 32 dw
  __shared__ float vns[32][33];

  v8f acc = {};
  for (int kc = 0; kc < KC; kc += 32) {
    if (tid < 32) {
      tdm_load_2d(red_w + (size_t)ob * KC + kc, &rws[0][0],
                  32, 64, KC, OUTD, KC, 4);
      tdm_load_2d(vnorm + kc, &vns[0][0],
                  32, 32, KC, NN, KC, 4);
      __builtin_amdgcn_s_wait_tensorcnt((short)0);
    }
    __syncthreads();
    #pragma unroll
    for (int ts = 0; ts < 32; ts += 4) {
      v2f a, b;
      a.x = vns[mt + r16][ts + 2*hi];            // A[M=n][Kdim=kc]
      a.y = vns[mt + r16][ts + 2*hi + 1];
      b.x = rws[ot + r16][ts + 2*hi];            // B[Kdim=kc][N=o] = red_w[o][kc]
      b.y = rws[ot + r16][ts + 2*hi + 1];
      acc = wmma4(a, b, acc);
    }
    __syncthreads();
  }
  V8 rr; rr.v = acc;
  #pragma unroll
  for (int i = 0; i < 8; ++i) {
    const int m = mt + i + 8*hi;
    const int o = ob + ot + r16;
    y[(size_t)m * OUTD + o] = rr.f[i];
  }
}

// =====================================================================
// Kernel 5: LayerNorm over OUT=1024 per n row
// =====================================================================
__global__ __launch_bounds__(256) void k_ln(
    const float* __restrict__ y, const float* __restrict__ ln_w,
    const float* __restrict__ ln_b, float* __restrict__ out)
{
  const int n = blockIdx.x, tid = threadIdx.x;
  __shared__ float red[256];
  const float* row = y + (size_t)n * OUTD;
  float v[4];
  #pragma unroll
  for (int j = 0; j < 4; ++j) v[j] = row[tid + j * 256];
  red[tid] = v[0] + v[1] + v[2] + v[3];
  __syncthreads();
  for (int off = 128; off > 0; off >>= 1) {
    if (tid < off) red[tid] += red[tid + off];
    __syncthreads();
  }
  const float mu = red[0] * (1.f / OUTD);
  __syncthreads();
  float ss = 0.f;
  #pragma unroll
  for (int j = 0; j < 4; ++j) { const float d = v[j] - mu; ss += d * d; }
  red[tid] = ss;
  __syncthreads();
  for (int off = 128; off > 0; off >>= 1) {
    if (tid < off) red[tid] += red[tid + off];
    __syncthreads();
  }
  const float var  = red[0] * (1.f / OUTD);
  const float rstd = 1.f / sqrtf(var + EPSL);
  #pragma unroll
  for (int j = 0; j < 4; ++j) {
    const int o = tid + j * 256;
    out[(size_t)n * OUTD + o] = (v[j] - mu) * rstd * ln_w[o] + ln_b[o];
  }
}

// =====================================================================
// Host-side launcher
// =====================================================================
extern "C" void kernel_launch(void* const* d_in, const int* in_sizes, int n_in,
                              void* d_out, int out_size, void* d_ws, size_t ws_size,
                              hipStream_t stream) {
  const float* x         = (const float*)d_in[0];  // (32,512,300,4)
  const int*   mask      = (const int*)  d_in[1];  // (32,300)
  const float* centroids = (const float*)d_in[2];  // (64,512)
  const float* conv_w    = (const float*)d_in[3];  // (64,512)
  const float* red_w     = (const float*)d_in[4];  // (1024,32768)
  const float* ln_w      = (const float*)d_in[5];  // (1024,)
  const float* ln_b      = (const float*)d_in[6];  // (1024,)
  float* out = (float*)d_out;                      // (32,1024)

  // workspace layout (floats): ~14.2 MB
  float* ws   = (float*)d_ws;
  float* am   = ws;                                  // N*K*TR
  float* asum = am   + (size_t)NN * KK * TRR;        // N*K
  float* vlad = asum + (size_t)NN * KK;              // N*K*C (normalized in place)
  float* y    = vlad + (size_t)NN * KK * CC;         // N*OUT

  k_assign<<<dim3(TRR / 16, NN), 128, 0, stream>>>(x, mask, conv_w, am);
  k_asum  <<<dim3(KK, NN),       128, 0, stream>>>(am, asum);
  k_vlad  <<<dim3(CC / 128, NN), 256, 0, stream>>>(x, am, asum, centroids, vlad);
  k_norm  <<<NN,                 256, 0, stream>>>(vlad);
  k_reduce<<<OUTD / 64,          256, 0, stream>>>(vlad, red_w, y);
  k_ln    <<<NN,                 256, 0, stream>>>(y, ln_w, ln_b, out);
}